// Net_19095424598712
// MI455X (gfx1250) — compile-verified
//
#include <hip/hip_runtime.h>

// GIN (2 layers) + mean-pool + MLP head for gfx1250 (MI455X).
// fp32 WMMA (v_wmma_f32_16x16x4_f32) for all 128-wide GEMMs.
// v2: register-resident B fragments (64 VGPRs/lane), 8 row-strips per block,
// double-buffered LDS A staging -> steady-state inner loop is ds_load + wmma only.

typedef __attribute__((ext_vector_type(2))) float v2f;
typedef __attribute__((ext_vector_type(8))) float v8f;

#define DIMC 128
#define LDS_STRIDE 132   // 128 + 4 pad: avoids 16-way LDS bank conflicts
#define BN_EPS 1e-5f

// ---------------- elementwise utility kernels ----------------

__global__ __launch_bounds__(256) void k_copy4(float4* __restrict__ dst,
                                               const float4* __restrict__ src,
                                               long n4) {
  long i = (long)blockIdx.x * blockDim.x + threadIdx.x;
  if (i < n4) dst[i] = src[i];
}

__global__ __launch_bounds__(256) void k_zero(float* __restrict__ p, long n) {
  long i = (long)blockIdx.x * blockDim.x + threadIdx.x;
  if (i < n) p[i] = 0.0f;
}

// ---------------- edge scatter-add: agg[dst] += x[src] ----------------
// One wave per edge, each lane moves 4 contiguous channels (float4 gather,
// 4x global_atomic_add_f32 scatter).
__global__ __launch_bounds__(256) void k_scatter_add(const float* __restrict__ x,
                                                     float* __restrict__ agg,
                                                     const int* __restrict__ ei,
                                                     int nE) {
  long gid = (long)blockIdx.x * 256 + threadIdx.x;
  int e = (int)(gid >> 5);
  if (e >= nE) return;
  int lane = (int)(gid & 31);
  int s = ei[e];
  int d = ei[(long)nE + e];
  const float4 v = *(const float4*)(x + (long)s * DIMC + lane * 4);
  float* dp = agg + (long)d * DIMC + lane * 4;
  unsafeAtomicAdd(dp + 0, v.x);
  unsafeAtomicAdd(dp + 1, v.y);
  unsafeAtomicAdd(dp + 2, v.z);
  unsafeAtomicAdd(dp + 3, v.w);
}

// ---------------- GEMM: out = relu(A[M x 128] @ W[128 x 128] + bias) --------
// Block = 256 threads (8 waves); wave w owns column tile n0 = 16*w.
// Each block processes SPB row-strips of 16 rows. B fragments for the whole
// K=128 loop are preloaded once into registers (32 x v2f = 64 VGPRs/lane);
// per strip only the 16x128 A tile is restaged via double-buffered LDS.
// All branches are block-uniform -> EXEC all-ones at every WMMA.
template <int SPB>
__global__ __launch_bounds__(256) void k_gemm_bias_relu(const float* __restrict__ A,
                                                        const float* __restrict__ W,
                                                        const float* __restrict__ bias,
                                                        float* __restrict__ out,
                                                        int nStrips) {
  __shared__ float ldsA[2][16 * LDS_STRIDE];
  const int t = threadIdx.x;
  const int wave = t >> 5;
  const int lane = t & 31;
  const int half = lane >> 4;    // K-half for A/B fragments
  const int lrow = lane & 15;    // M for A-frag, N for B-frag and C column
  const int n0 = wave * 16;

  // Preload all B fragments (4x16 fp32 each): lane = N column, regs span K.
  v2f breg[DIMC / 4];
#pragma unroll
  for (int kk = 0; kk < DIMC / 4; ++kk) {
    const int ka = kk * 4 + 2 * half;
    breg[kk].x = W[(long)ka * DIMC + n0 + lrow];
    breg[kk].y = W[(long)(ka + 1) * DIMC + n0 + lrow];
  }
  const float bv = bias[n0 + lrow];

  const int srow = t >> 4;        // A-staging row   (0..15)
  const int scol = (t & 15) * 8;  // A-staging col   (0..120)

  for (int s = 0; s < SPB; ++s) {
    const int strip = blockIdx.x * SPB + s;   // scalar -> uniform control flow
    if (strip >= nStrips) break;
    const long m0 = (long)strip * 16;
    const int buf = s & 1;

    // Stage 16x128 A-strip into LDS (double-buffered, one barrier per strip).
    {
      const float4* src = (const float4*)(A + (m0 + srow) * DIMC + scol);
      float4 v0 = src[0];
      float4 v1 = src[1];
      *(float4*)(&ldsA[buf][srow * LDS_STRIDE + scol]) = v0;
      *(float4*)(&ldsA[buf][srow * LDS_STRIDE + scol + 4]) = v1;
    }
    __syncthreads();

    v8f acc = {0.f, 0.f, 0.f, 0.f, 0.f, 0.f, 0.f, 0.f};
#pragma unroll
    for (int kk = 0; kk < DIMC / 4; ++kk) {
      // A fragment (16x4 fp32): lanes 0-15 K={k,k+1}, lanes 16-31 K={k+2,k+3}
      v2f a = *(const v2f*)(&ldsA[buf][lrow * LDS_STRIDE + kk * 4 + 2 * half]);
      acc = __builtin_amdgcn_wmma_f32_16x16x4_f32(false, a, false, breg[kk],
                                                  (short)0, acc, false, false);
    }

#pragma unroll
    for (int j = 0; j < 8; ++j) {
      long row = m0 + j + half * 8;    // C VGPR j -> M=j (lanes 0-15) / j+8
      float v = acc[j] + bv;
      v = v > 0.0f ? v : 0.0f;
      out[row * DIMC + n0 + lrow] = v;
    }
  }
}

// ---------------- BatchNorm statistics (pass 1) ----------------
// thread = channel (coalesced column reads), fp32 atomic partial reduction.
__global__ __launch_bounds__(128) void k_bn_stats(const float* __restrict__ h,
                                                  float* __restrict__ sum,
                                                  float* __restrict__ sq,
                                                  int M) {
  int c = threadIdx.x;
  int chunk = (M + gridDim.x - 1) / gridDim.x;
  int r0 = blockIdx.x * chunk;
  int r1 = r0 + chunk;
  if (r1 > M) r1 = M;
  float s = 0.0f, q = 0.0f;
  for (int r = r0; r < r1; ++r) {
    float v = h[(long)r * DIMC + c];
    s += v;
    q = fmaf(v, v, q);
  }
  unsafeAtomicAdd(&sum[c], s);
  unsafeAtomicAdd(&sq[c], q);
}

// ---------------- BatchNorm apply (pass 2, in place) ----------------
__global__ __launch_bounds__(256) void k_bn_apply(float* __restrict__ h,
                                                  const float* __restrict__ sum,
                                                  const float* __restrict__ sq,
                                                  const float* __restrict__ g,
                                                  const float* __restrict__ be,
                                                  int M) {
  long i = (long)blockIdx.x * 256 + threadIdx.x;
  long n4 = (long)M * (DIMC / 4);
  if (i >= n4) return;
  int c0 = (int)((i * 4) & (DIMC - 1));
  float invM = 1.0f / (float)M;
  float4 v = ((float4*)h)[i];
  float* vp = (float*)&v;
#pragma unroll
  for (int j = 0; j < 4; ++j) {
    int c = c0 + j;
    float mean = sum[c] * invM;
    float var = fmaf(-mean, mean, sq[c] * invM);
    vp[j] = (vp[j] - mean) * rsqrtf(var + BN_EPS) * g[c] + be[c];
  }
  ((float4*)h)[i] = v;
}

// ---------------- global mean pool (sum + count) ----------------
__global__ __launch_bounds__(256) void k_pool(const float* __restrict__ h,
                                              const int* __restrict__ batch,
                                              float* __restrict__ pooled,
                                              float* __restrict__ counts,
                                              int nN) {
  long gid = (long)blockIdx.x * 256 + threadIdx.x;
  int n = (int)(gid >> 5);
  if (n >= nN) return;
  int lane = (int)(gid & 31);
  int g = batch[n];
  const float4 v = *(const float4*)(h + (long)n * DIMC + lane * 4);
  float* dp = pooled + (long)g * DIMC + lane * 4;
  unsafeAtomicAdd(dp + 0, v.x);
  unsafeAtomicAdd(dp + 1, v.y);
  unsafeAtomicAdd(dp + 2, v.z);
  unsafeAtomicAdd(dp + 3, v.w);
  if (lane == 0) unsafeAtomicAdd(&counts[g], 1.0f);
}

__global__ __launch_bounds__(256) void k_pool_div(float* __restrict__ pooled,
                                                  const float* __restrict__ counts,
                                                  int nG) {
  int i = blockIdx.x * 256 + threadIdx.x;
  if (i < nG * DIMC) pooled[i] /= fmaxf(counts[i >> 7], 1.0f);
}

// ---------------- head: logits = hidden @ lw2 + lb2, then log_softmax -------
__global__ __launch_bounds__(64) void k_head_out(const float* __restrict__ hid,
                                                 const float* __restrict__ w,
                                                 const float* __restrict__ b,
                                                 float* __restrict__ out,
                                                 int nG) {
  int g = blockIdx.x * 64 + threadIdx.x;
  if (g >= nG) return;
  float logit[10];
#pragma unroll
  for (int o = 0; o < 10; ++o) logit[o] = b[o];
  for (int k = 0; k < DIMC; ++k) {
    float hv = hid[(long)g * DIMC + k];
#pragma unroll
    for (int o = 0; o < 10; ++o) logit[o] = fmaf(hv, w[k * 10 + o], logit[o]);
  }
  float m = logit[0];
#pragma unroll
  for (int o = 1; o < 10; ++o) m = fmaxf(m, logit[o]);
  float s = 0.0f;
#pragma unroll
  for (int o = 0; o < 10; ++o) s += expf(logit[o] - m);
  float lse = m + logf(s);
#pragma unroll
  for (int o = 0; o < 10; ++o) out[(long)g * 10 + o] = logit[o] - lse;
}

// ---------------- host orchestration ----------------

extern "C" void kernel_launch(void* const* d_in, const int* in_sizes, int n_in,
                              void* d_out, int out_size, void* d_ws, size_t ws_size,
                              hipStream_t stream) {
  const float* x   = (const float*)d_in[0];
  const int* ei    = (const int*)d_in[1];
  const int* batch = (const int*)d_in[2];
  const float* w11 = (const float*)d_in[3];
  const float* b11 = (const float*)d_in[4];
  const float* w12 = (const float*)d_in[5];
  const float* b12 = (const float*)d_in[6];
  const float* g1  = (const float*)d_in[7];
  const float* be1 = (const float*)d_in[8];
  const float* w21 = (const float*)d_in[9];
  const float* b21 = (const float*)d_in[10];
  const float* w22 = (const float*)d_in[11];
  const float* b22 = (const float*)d_in[12];
  const float* g2  = (const float*)d_in[13];
  const float* be2 = (const float*)d_in[14];
  const float* lw1 = (const float*)d_in[15];
  const float* lb1 = (const float*)d_in[16];
  const float* lw2 = (const float*)d_in[17];
  const float* lb2 = (const float*)d_in[18];
  float* out = (float*)d_out;

  const int nN = in_sizes[0] / DIMC;  // 100000 nodes
  const int nE = in_sizes[1] / 2;     // 1600000 edges
  const int nG = out_size / 10;       // 512 graphs

  // workspace carve-out (fp32): two ping-pong node buffers + small scratch
  float* ws     = (float*)d_ws;
  float* bufA   = ws;
  float* bufB   = bufA + (long)nN * DIMC;
  float* bnsum  = bufB + (long)nN * DIMC;
  float* bnsq   = bnsum + DIMC;
  float* pooled = bnsq + DIMC;
  float* counts = pooled + (long)nG * DIMC;
  float* hidden = counts + nG;

  const long n4       = (long)nN * (DIMC / 4);
  const int cpBlocks  = (int)((n4 + 255) / 256);
  const int scBlocks  = (int)(((long)nE * 32 + 255) / 256);
  const int poolBlk   = (int)(((long)nN * 32 + 255) / 256);

  constexpr int SPB   = 8;                        // row-strips per block
  const int nStripsN  = nN / 16;                  // 6250
  const int nStripsG  = nG / 16;                  // 32
  const int gemmBlkN  = (nStripsN + SPB - 1) / SPB;
  const int gemmBlkG  = (nStripsG + SPB - 1) / SPB;

  // ---- GIN layer 1: h_pre = x + scatter(x) ----
  k_copy4<<<cpBlocks, 256, 0, stream>>>((float4*)bufA, (const float4*)x, n4);
  k_scatter_add<<<scBlocks, 256, 0, stream>>>(x, bufA, ei, nE);
  k_gemm_bias_relu<SPB><<<gemmBlkN, 256, 0, stream>>>(bufA, w11, b11, bufB, nStripsN);
  k_gemm_bias_relu<SPB><<<gemmBlkN, 256, 0, stream>>>(bufB, w12, b12, bufA, nStripsN);
  k_zero<<<1, 256, 0, stream>>>(bnsum, 2 * DIMC);
  k_bn_stats<<<256, 128, 0, stream>>>(bufA, bnsum, bnsq, nN);
  k_bn_apply<<<cpBlocks, 256, 0, stream>>>(bufA, bnsum, bnsq, g1, be1, nN);
  // h1 now in bufA

  // ---- GIN layer 2 ----
  k_copy4<<<cpBlocks, 256, 0, stream>>>((float4*)bufB, (const float4*)bufA, n4);
  k_scatter_add<<<scBlocks, 256, 0, stream>>>(bufA, bufB, ei, nE);
  k_gemm_bias_relu<SPB><<<gemmBlkN, 256, 0, stream>>>(bufB, w21, b21, bufA, nStripsN);
  k_gemm_bias_relu<SPB><<<gemmBlkN, 256, 0, stream>>>(bufA, w22, b22, bufB, nStripsN);
  k_zero<<<1, 256, 0, stream>>>(bnsum, 2 * DIMC);
  k_bn_stats<<<256, 128, 0, stream>>>(bufB, bnsum, bnsq, nN);
  k_bn_apply<<<cpBlocks, 256, 0, stream>>>(bufB, bnsum, bnsq, g2, be2, nN);
  // h2 now in bufB

  // ---- mean pool + head ----
  k_zero<<<(int)(((long)nG * DIMC + nG + 255) / 256), 256, 0, stream>>>(
      pooled, (long)nG * DIMC + nG);
  k_pool<<<poolBlk, 256, 0, stream>>>(bufB, batch, pooled, counts, nN);
  k_pool_div<<<(nG * DIMC + 255) / 256, 256, 0, stream>>>(pooled, counts, nG);
  k_gemm_bias_relu<SPB><<<gemmBlkG, 256, 0, stream>>>(pooled, lw1, lb1, hidden, nStripsG);
  k_head_out<<<(nG + 63) / 64, 64, 0, stream>>>(hidden, lw2, lb2, out, nG);
}